// GCU_36490042147210
// MI455X (gfx1250) — compile-verified
//
#include <hip/hip_runtime.h>
#include <hip/hip_bf16.h>
#include <stdint.h>

#define NNODES 50000
#define NEDGES 800000
#define M_ROWS (NEDGES + NNODES) /* 850000 */
#define EPSBN 1e-5f
#define WAVES 4
#define TPW 8
#define TPWF 4

typedef float v2f __attribute__((ext_vector_type(2)));
typedef float v8f __attribute__((ext_vector_type(8)));

__device__ __forceinline__ v8f wmma4(v2f a, v2f b, v8f c) {
  // D = A(16x4) * B(4x16) + C, fp32
  return __builtin_amdgcn_wmma_f32_16x16x4_f32(false, a, false, b, (short)0, c,
                                               false, false);
}

__device__ __forceinline__ unsigned enc_key(float v) {
  unsigned b = __float_as_uint(v);
  return b ^ (unsigned)(((int)b >> 31) | 0x80000000);
}
__device__ __forceinline__ float dec_key(unsigned k) {
  unsigned b = (k & 0x80000000u) ? (k ^ 0x80000000u) : ~k;
  return __uint_as_float(b);
}

// ---------------------------------------------------------------------------
// init: zero segment-max keys + stats area
// ---------------------------------------------------------------------------
__global__ void k_init(unsigned* __restrict__ keys, float* __restrict__ stats) {
  long idx = (long)blockIdx.x * blockDim.x + threadIdx.x;
  if (idx < (long)NNODES * 128) keys[idx] = 0u;  // below enc(-inf)
  if (idx < 1536) stats[idx] = 0.f;
}

// ---------------------------------------------------------------------------
// conv layer 1: h = relu([xi, xj-xi] @ W1 + b1), accumulate sum/sumsq
// ---------------------------------------------------------------------------
__global__ __launch_bounds__(128) void k_edge1(
    const float* __restrict__ x, const int* __restrict__ ei,
    const float* __restrict__ W1, const float* __restrict__ b1,
    float* __restrict__ h, float* __restrict__ stats) {
  __shared__ __align__(16) float wfrag[32 * 4 * 64];  // 32KB: B frags (kt,nt)
  __shared__ float red[128];
  const int tid = threadIdx.x;

  // stage W1[128][64] into WMMA B-fragment order
  for (int idx = tid; idx < 128 * 64; idx += 128) {
    int k = idx >> 6, n = idx & 63;
    int kt = k >> 2, kk = k & 3;
    int nt = n >> 4, col = n & 15;
    wfrag[((kt * 4 + nt) << 6) + ((col + ((kk >> 1) << 4)) << 1) + (kk & 1)] =
        W1[k * 64 + n];
  }
  red[tid] = 0.f;
  __syncthreads();

  const int lane = tid & 31;
  const int w = tid >> 5;
  const int row = lane & 15;
  const int hs = lane >> 4;
  const int koff = hs << 1;
  const v8f z8 = {0, 0, 0, 0, 0, 0, 0, 0};

  float psum[4] = {0, 0, 0, 0}, psq[4] = {0, 0, 0, 0};

#pragma unroll 1
  for (int t = 0; t < TPW; ++t) {
    long base = ((long)blockIdx.x * (WAVES * TPW) + w * TPW + t) * 16;
    long eA = base + row;
    if (eA >= M_ROWS) eA = M_ROWS - 1;
    int srcN = (eA < NEDGES) ? ei[eA] : (int)(eA - NEDGES);
    int dstN = (eA < NEDGES) ? ei[NEDGES + eA] : (int)(eA - NEDGES);
    const float2* xd2 = (const float2*)(x + (long)dstN * 64);
    const float2* xs2 = (const float2*)(x + (long)srcN * 64);

    v8f acc[4] = {z8, z8, z8, z8};
#pragma unroll
    for (int kt = 0; kt < 32; ++kt) {
      v2f a;
      if (kt < 16) {  // xi columns 0..63
        float2 d = xd2[(4 * kt + koff) >> 1];
        a.x = d.x; a.y = d.y;
      } else {        // (xj - xi) columns 64..127
        int c2 = (4 * (kt - 16) + koff) >> 1;
        float2 s = xs2[c2], d = xd2[c2];
        a.x = s.x - d.x; a.y = s.y - d.y;
      }
#pragma unroll
      for (int nt = 0; nt < 4; ++nt) {
        v2f b = *(const v2f*)(wfrag + ((kt * 4 + nt) << 6) + (lane << 1));
        acc[nt] = wmma4(a, b, acc[nt]);
      }
    }
#pragma unroll
    for (int nt = 0; nt < 4; ++nt) {
      int n = nt * 16 + row;
      float bias = b1[n];
#pragma unroll
      for (int j = 0; j < 8; ++j) {
        long gr = base + hs * 8 + j;
        if (gr < M_ROWS) {
          float z = acc[nt][j] + bias;
          z = z > 0.f ? z : 0.f;
          h[gr * 64 + n] = z;
          psum[nt] += z;
          psq[nt] += z * z;
        }
      }
    }
  }
#pragma unroll
  for (int nt = 0; nt < 4; ++nt) {
    int n = nt * 16 + row;
    atomicAdd(&red[n], psum[nt]);
    atomicAdd(&red[64 + n], psq[nt]);
  }
  __syncthreads();
  atomicAdd(&stats[tid], red[tid]);
}

// ---------------------------------------------------------------------------
// conv layer 2 (in place): h = relu(bn1(h) @ W2 + b2)
// ---------------------------------------------------------------------------
__global__ __launch_bounds__(128) void k_edge2(
    float* __restrict__ h, const float* __restrict__ W2,
    const float* __restrict__ b2, const float* __restrict__ st1,
    float* __restrict__ stats2) {
  __shared__ __align__(16) float wfrag[16 * 4 * 64];  // 16KB
  __shared__ float red[128];
  const int tid = threadIdx.x;
  for (int idx = tid; idx < 64 * 64; idx += 128) {
    int k = idx >> 6, n = idx & 63;
    int kt = k >> 2, kk = k & 3;
    int nt = n >> 4, col = n & 15;
    wfrag[((kt * 4 + nt) << 6) + ((col + ((kk >> 1) << 4)) << 1) + (kk & 1)] =
        W2[k * 64 + n];
  }
  red[tid] = 0.f;
  __syncthreads();

  const float* s1 = st1 + 128;
  const float* t1 = st1 + 192;
  const int lane = tid & 31;
  const int w = tid >> 5;
  const int row = lane & 15;
  const int hs = lane >> 4;
  const int koff = hs << 1;
  const v8f z8 = {0, 0, 0, 0, 0, 0, 0, 0};

  float psum[4] = {0, 0, 0, 0}, psq[4] = {0, 0, 0, 0};

#pragma unroll 1
  for (int t = 0; t < TPW; ++t) {
    long base = ((long)blockIdx.x * (WAVES * TPW) + w * TPW + t) * 16;
    long eA = base + row;
    if (eA >= M_ROWS) eA = M_ROWS - 1;
    const float2* hr = (const float2*)(h + eA * 64);

    v8f acc[4] = {z8, z8, z8, z8};
#pragma unroll
    for (int kt = 0; kt < 16; ++kt) {
      int c = 4 * kt + koff;
      float2 hv = hr[c >> 1];
      v2f a;
      a.x = hv.x * s1[c] + t1[c];
      a.y = hv.y * s1[c + 1] + t1[c + 1];
#pragma unroll
      for (int nt = 0; nt < 4; ++nt) {
        v2f b = *(const v2f*)(wfrag + ((kt * 4 + nt) << 6) + (lane << 1));
        acc[nt] = wmma4(a, b, acc[nt]);
      }
    }
#pragma unroll
    for (int nt = 0; nt < 4; ++nt) {
      int n = nt * 16 + row;
      float bias = b2[n];
#pragma unroll
      for (int j = 0; j < 8; ++j) {
        long gr = base + hs * 8 + j;
        if (gr < M_ROWS) {
          float z = acc[nt][j] + bias;
          z = z > 0.f ? z : 0.f;
          h[gr * 64 + n] = z;  // in place: store data depends on all A loads
          psum[nt] += z;
          psq[nt] += z * z;
        }
      }
    }
  }
#pragma unroll
  for (int nt = 0; nt < 4; ++nt) {
    int n = nt * 16 + row;
    atomicAdd(&red[n], psum[nt]);
    atomicAdd(&red[64 + n], psq[nt]);
  }
  __syncthreads();
  atomicAdd(&stats2[tid], red[tid]);
}

// ---------------------------------------------------------------------------
// BN stats -> affine (s,t) per channel; slot = [sum(C), sumsq(C), s(C), t(C)]
// ---------------------------------------------------------------------------
__global__ void k_stats(float* __restrict__ slot, const float* __restrict__ g,
                        const float* __restrict__ be, int C, float invCnt) {
  int c = threadIdx.x;
  if (c < C) {
    float mean = slot[c] * invCnt;
    float var = slot[C + c] * invCnt - mean * mean;
    float s = g[c] * rsqrtf(var + EPSBN);
    slot[2 * C + c] = s;
    slot[3 * C + c] = be[c] - mean * s;
  }
}

// ---------------------------------------------------------------------------
// segment max: keys[dst, colOff+c] = max(enc(bn2(h[e,c])))
// ---------------------------------------------------------------------------
__global__ void k_max(const float* __restrict__ h, const int* __restrict__ ei,
                      const float* __restrict__ st2,
                      unsigned* __restrict__ keys, int colOff) {
  long idx = (long)blockIdx.x * blockDim.x + threadIdx.x;
  if (idx >= (long)M_ROWS * 16) return;
  long e = idx >> 4;
  int g = (int)(idx & 15);
  int dstN = (e < NEDGES) ? ei[NEDGES + e] : (int)(e - NEDGES);
  const float4 hv = *(const float4*)(h + e * 64 + g * 4);
  const float4 sv = *(const float4*)(st2 + 128 + g * 4);
  const float4 tv = *(const float4*)(st2 + 192 + g * 4);
  unsigned* kp = keys + (long)dstN * 128 + colOff + g * 4;
  atomicMax(&kp[0], enc_key(hv.x * sv.x + tv.x));
  atomicMax(&kp[1], enc_key(hv.y * sv.y + tv.y));
  atomicMax(&kp[2], enc_key(hv.z * sv.z + tv.z));
  atomicMax(&kp[3], enc_key(hv.w * sv.w + tv.w));
}

// ---------------------------------------------------------------------------
// final: z = relu(dec(keys) @ W_out + b_out), accumulate stats (half of cols
// per blockIdx.y)
// ---------------------------------------------------------------------------
__global__ __launch_bounds__(128) void k_f1(const unsigned* __restrict__ keys,
                                            const float* __restrict__ Wout,
                                            const float* __restrict__ bout,
                                            float* __restrict__ z,
                                            float* __restrict__ statsF) {
  __shared__ __align__(16) float wfrag[32 * 4 * 64];  // 32KB (64 output cols)
  __shared__ float red[128];
  const int tid = threadIdx.x;
  const int yy = blockIdx.y;  // column half
  for (int idx = tid; idx < 128 * 64; idx += 128) {
    int k = idx >> 6, nl = idx & 63;
    int kt = k >> 2, kk = k & 3;
    int nt = nl >> 4, col = nl & 15;
    wfrag[((kt * 4 + nt) << 6) + ((col + ((kk >> 1) << 4)) << 1) + (kk & 1)] =
        Wout[k * 128 + yy * 64 + nl];
  }
  red[tid] = 0.f;
  __syncthreads();

  const int lane = tid & 31;
  const int w = tid >> 5;
  const int row = lane & 15;
  const int hs = lane >> 4;
  const int koff = hs << 1;
  const v8f z8 = {0, 0, 0, 0, 0, 0, 0, 0};

  float psum[4] = {0, 0, 0, 0}, psq[4] = {0, 0, 0, 0};

#pragma unroll 1
  for (int t = 0; t < TPWF; ++t) {
    long base = ((long)blockIdx.x * (WAVES * TPWF) + w * TPWF + t) * 16;
    long r = base + row;
    if (r >= NNODES) r = NNODES - 1;
    const unsigned* kp = keys + r * 128;

    v8f acc[4] = {z8, z8, z8, z8};
#pragma unroll
    for (int kt = 0; kt < 32; ++kt) {
      uint2 kv = *(const uint2*)(kp + 4 * kt + koff);
      v2f a;
      a.x = dec_key(kv.x);
      a.y = dec_key(kv.y);
#pragma unroll
      for (int nt = 0; nt < 4; ++nt) {
        v2f b = *(const v2f*)(wfrag + ((kt * 4 + nt) << 6) + (lane << 1));
        acc[nt] = wmma4(a, b, acc[nt]);
      }
    }
#pragma unroll
    for (int nt = 0; nt < 4; ++nt) {
      int nl = nt * 16 + row;
      int n = yy * 64 + nl;
      float bias = bout[n];
#pragma unroll
      for (int j = 0; j < 8; ++j) {
        long gr = base + hs * 8 + j;
        if (gr < NNODES) {
          float zz = acc[nt][j] + bias;
          zz = zz > 0.f ? zz : 0.f;
          z[gr * 128 + n] = zz;
          psum[nt] += zz;
          psq[nt] += zz * zz;
        }
      }
    }
  }
#pragma unroll
  for (int nt = 0; nt < 4; ++nt) {
    int nl = nt * 16 + row;
    atomicAdd(&red[nl], psum[nt]);
    atomicAdd(&red[64 + nl], psq[nt]);
  }
  __syncthreads();
  if (tid < 64)
    atomicAdd(&statsF[yy * 64 + tid], red[tid]);
  else
    atomicAdd(&statsF[128 + yy * 64 + (tid - 64)], red[tid]);
}

__global__ void k_f2(const float* __restrict__ z,
                     const float* __restrict__ statsF,
                     float* __restrict__ out) {
  long idx = (long)blockIdx.x * blockDim.x + threadIdx.x;
  if (idx < (long)NNODES * 128) {
    int n = (int)(idx & 127);
    out[idx] = z[idx] * statsF[256 + n] + statsF[384 + n];
  }
}

// ---------------------------------------------------------------------------
extern "C" void kernel_launch(void* const* d_in, const int* in_sizes, int n_in,
                              void* d_out, int out_size, void* d_ws,
                              size_t ws_size, hipStream_t stream) {
  const float* x = (const float*)d_in[0];
  const int* ei[2] = {(const int*)d_in[1], (const int*)d_in[2]};
  // params0: d_in[3..10] = W1,b1,g1,be1,W2,b2,g2,be2 ; params1: d_in[11..18]
  const float *W1[2], *b1[2], *g1[2], *be1[2], *W2[2], *b2[2], *g2[2], *be2[2];
  for (int c = 0; c < 2; ++c) {
    int o = 3 + 8 * c;
    W1[c] = (const float*)d_in[o + 0];
    b1[c] = (const float*)d_in[o + 1];
    g1[c] = (const float*)d_in[o + 2];
    be1[c] = (const float*)d_in[o + 3];
    W2[c] = (const float*)d_in[o + 4];
    b2[c] = (const float*)d_in[o + 5];
    g2[c] = (const float*)d_in[o + 6];
    be2[c] = (const float*)d_in[o + 7];
  }
  const float* Wout = (const float*)d_in[19];
  const float* bout = (const float*)d_in[20];
  const float* gout = (const float*)d_in[21];
  const float* beout = (const float*)d_in[22];

  char* ws = (char*)d_ws;
  float* h = (float*)ws;  // M_ROWS*64 f32 (217.6 MB)
  size_t off = (size_t)M_ROWS * 64 * 4;
  unsigned* keys = (unsigned*)(ws + off);  // N*128 u32 (25.6 MB)
  off += (size_t)NNODES * 128 * 4;
  float* zbuf = (float*)(ws + off);  // N*128 f32 (25.6 MB)
  off += (size_t)NNODES * 128 * 4;
  float* stats = (float*)(ws + off);  // 1536 f32

  {
    long n = (long)NNODES * 128;
    k_init<<<(int)((n + 255) / 256), 256, 0, stream>>>(keys, stats);
  }

  const int tiles1 = (M_ROWS + 15) / 16;  // 53125
  const int grid1 = (tiles1 + WAVES * TPW - 1) / (WAVES * TPW);
  const float invM = 1.0f / (float)M_ROWS;

  for (int c = 0; c < 2; ++c) {
    float* slot1 = stats + c * 512;
    float* slot2 = slot1 + 256;
    k_edge1<<<grid1, 128, 0, stream>>>(x, ei[c], W1[c], b1[c], h, slot1);
    k_stats<<<1, 128, 0, stream>>>(slot1, g1[c], be1[c], 64, invM);
    k_edge2<<<grid1, 128, 0, stream>>>(h, W2[c], b2[c], slot1, slot2);
    k_stats<<<1, 128, 0, stream>>>(slot2, g2[c], be2[c], 64, invM);
    long tm = (long)M_ROWS * 16;
    k_max<<<(int)((tm + 255) / 256), 256, 0, stream>>>(h, ei[c], slot2, keys,
                                                       c * 64);
  }

  float* fslot = stats + 1024;
  const int tilesN = (NNODES + 15) / 16;  // 3125
  const int gridF = (tilesN + WAVES * TPWF - 1) / (WAVES * TPWF);
  dim3 gf(gridF, 2, 1);
  k_f1<<<gf, 128, 0, stream>>>(keys, Wout, bout, zbuf, fslot);
  k_stats<<<1, 128, 0, stream>>>(fslot, gout, beout, 128, 1.0f / NNODES);
  {
    long n = (long)NNODES * 128;
    k_f2<<<(int)((n + 255) / 256), 256, 0, stream>>>(zbuf, fslot,
                                                     (float*)d_out);
  }
}